// NestedAttention_40862318854317
// MI455X (gfx1250) — compile-verified
//
#include <hip/hip_runtime.h>
#include <stdint.h>

// NestedAttention fused kernel for gfx1250 (MI455X), f32 WMMA path, TM=32.
//
// B=16, N=1024, DIM=768, HEADS=12, HEAD_DIM=64, nested widths 96/192/384/768.
// One workgroup = 32 tokens, 16 wave32s. Fully fused:
//   masked-x (streamed K-panels, async copy to LDS) -> qkv GEMM (f32 WMMA,
//   3 col tiles x 2 row tiles per wave) -> per-token head-axis attention
//   (wave=head, lane=token) -> proj GEMM + bias + nested output mask.
// LDS: 2x[32x68] x-panels (17KB) + [32x2308] qkv (295KB) ~ 305KB < 320KB WGP LDS.
// Weight L2 traffic: 9.3MB x 512 WGs ~ 4.75GB (halved vs TM=16); HBM ~ 250MB.

typedef __attribute__((ext_vector_type(2))) float v2f;
typedef __attribute__((ext_vector_type(8))) float v8f;
typedef int async_b128 __attribute__((vector_size(16)));   // matches builtin proto v4i

#define DIMX   768
#define NHEAD  12
#define HDIM   64
#define TM     32          // tokens per workgroup (two 16-row WMMA tiles)
#define KP     64          // K-panel depth streamed through LDS
#define XPSTR  68          // 64 + 4 pad words -> row stride % 64 banks == 4
#define QSTR   2308        // 2304 + 4 pad words
#define NTHR   512         // 16 wave32s
#define ATTN_SCALE 0.125f  // 64^-0.5

#if defined(__has_builtin)
#if __has_builtin(__builtin_amdgcn_global_load_async_to_lds_b128)
#define USE_ASYNC_COPY 1
#endif
#endif

__device__ __forceinline__ void stage16(const float* gsrc, float* ldst) {
#ifdef USE_ASYNC_COPY
    // ASYNCcnt-tracked copy, no VGPR round trip. Generic LDS pointer's low 32
    // bits are the LDS offset; generic global pointer equals the AS1 address.
    auto gp = (__attribute__((address_space(1))) async_b128*)(uintptr_t)gsrc;
    auto lp = (__attribute__((address_space(3))) async_b128*)(uint32_t)(uintptr_t)ldst;
    __builtin_amdgcn_global_load_async_to_lds_b128(gp, lp, 0, 0);
#else
    *reinterpret_cast<float4*>(ldst) = *reinterpret_cast<const float4*>(gsrc);
#endif
}

__device__ __forceinline__ void wait_stage() {
#ifdef USE_ASYNC_COPY
#if __has_builtin(__builtin_amdgcn_s_wait_asynccnt)
    __builtin_amdgcn_s_wait_asynccnt(0);
#else
    asm volatile("s_wait_asynccnt 0x0" ::: "memory");
#endif
#endif
}

__global__ __launch_bounds__(NTHR, 1)
void nested_attn_fused2_kernel(const float* __restrict__ x,
                               const int*   __restrict__ emask,
                               const float* __restrict__ qkv_w,   // [2304,768] row-major
                               const float* __restrict__ proj_w,  // [768,768]  row-major
                               const float* __restrict__ proj_b,  // [768]
                               float*       __restrict__ out)     // [16384,768]
{
    extern __shared__ float smem[];
    float* xp  = smem;                   // [2][TM][XPSTR] double-buffered x K-panels
    float* qkv = smem + 2 * TM * XPSTR;  // [TM][QSTR] per-token q|k|v, q later = attn x
    __shared__ int de[TM];

    const int tid = threadIdx.x;
    const int tb  = blockIdx.x * TM;

    if (tid < TM) de[tid] = 96 << emask[tb + tid];   // 768 >> (3 - e)

    const int lane = tid & 31;
    const int wv   = tid >> 5;             // wave 0..15
    const int mrow = lane & 15;            // A row within tile / B col (out feature)
    const int koff = (lane >> 4) << 1;     // per-lane K sub-offset: 0 or 2
    const int hi8  = (lane >> 4) << 3;     // C row offset: 0 or 8

    // staging coords: one 16B chunk per thread, 32 tokens x 16 chunks = 512
    const int st_t = tid >> 4;
    const int st_f = (tid & 15) << 2;
    const float* xsrc = x + (size_t)(tb + st_t) * DIMX + st_f;
    float* lbuf0 = xp + st_t * XPSTR + st_f;
    float* lbuf1 = lbuf0 + TM * XPSTR;

    __syncthreads();                       // publish de[]
    const int d0 = de[mrow], d1 = de[mrow + 16];

    // ================= Phase 2: qkv GEMM (144 col tiles = 3 groups x 16 waves x 3) =====
    for (int grp = 0; grp < 3; ++grp) {
        const int tilebase = grp * 48 + wv * 3;
        const float* bp[3];
#pragma unroll
        for (int jj = 0; jj < 3; ++jj)
            bp[jj] = qkv_w + (size_t)((tilebase + jj) * 16 + mrow) * DIMX;

        v8f c[3][2] = {};

        stage16(xsrc, lbuf0);              // panel 0 -> buffer 0
        wait_stage();
        __syncthreads();

        for (int p = 0; p < 12; ++p) {
            if (p < 11) stage16(xsrc + (p + 1) * KP, (p & 1) ? lbuf0 : lbuf1);
            const float* xb = smem + (p & 1) * TM * XPSTR;
            const int kbase = p * KP;
#pragma unroll 4
            for (int kb = 0; kb < KP; kb += 4) {
                const int kg = kbase + kb + koff;          // global K of this fragment
                v2f a0 = *reinterpret_cast<const v2f*>(&xb[mrow * XPSTR + kb + koff]);
                v2f a1 = *reinterpret_cast<const v2f*>(&xb[(mrow + 16) * XPSTR + kb + koff]);
                if (kg >= d0) { a0.x = 0.0f; a0.y = 0.0f; }   // nested feature mask
                if (kg >= d1) { a1.x = 0.0f; a1.y = 0.0f; }   // (pair-uniform: de%4==0)
#pragma unroll
                for (int jj = 0; jj < 3; ++jj) {
                    v2f b = *reinterpret_cast<const v2f*>(&bp[jj][kg]);
                    c[jj][0] = __builtin_amdgcn_wmma_f32_16x16x4_f32(
                        false, a0, false, b, (short)0, c[jj][0], false, false);
                    c[jj][1] = __builtin_amdgcn_wmma_f32_16x16x4_f32(
                        false, a1, false, b, (short)0, c[jj][1], false, false);
                }
            }
            wait_stage();
            __syncthreads();
        }
#pragma unroll
        for (int jj = 0; jj < 3; ++jj) {
            const int cb = (tilebase + jj) * 16 + mrow;
#pragma unroll
            for (int rt = 0; rt < 2; ++rt)
#pragma unroll
                for (int i = 0; i < 8; ++i)
                    qkv[(rt * 16 + hi8 + i) * QSTR + cb] = c[jj][rt][i];
        }
    }
    __syncthreads();

    // ================= Phase 3: per-token attention over HEADS axis =====================
    // wave = head (12 active), lane = token (32). 12x12 scores, softmax, 12x64 mix.
    float sw[NHEAD];
    const int at = lane;                   // token within tile
    const int ah = wv;                     // head
    const bool attn_on = (wv < NHEAD);
    if (attn_on) {
        const float* tok = &qkv[at * QSTR];
        const float* qh  = tok + ah * HDIM;
        float mx = -3.0e38f;
#pragma unroll
        for (int ss = 0; ss < NHEAD; ++ss) {
            const float* kr = tok + DIMX + ss * HDIM;
            float acc = 0.0f;
#pragma unroll 8
            for (int d = 0; d < HDIM; ++d) acc = fmaf(qh[d], kr[d], acc);
            sw[ss] = acc * ATTN_SCALE;
            mx = fmaxf(mx, sw[ss]);
        }
        float sum = 0.0f;
#pragma unroll
        for (int ss = 0; ss < NHEAD; ++ss) { sw[ss] = __expf(sw[ss] - mx); sum += sw[ss]; }
        const float inv = 1.0f / sum;
#pragma unroll
        for (int ss = 0; ss < NHEAD; ++ss) sw[ss] *= inv;
    }
    __syncthreads();                       // all q/k reads done before q is overwritten
    if (attn_on) {
        const float* tokv = &qkv[at * QSTR + 2 * DIMX];
        float* xo = &qkv[at * QSTR + ah * HDIM];   // attended x -> q region
        for (int d = 0; d < HDIM; ++d) {
            float acc = 0.0f;
#pragma unroll
            for (int ss = 0; ss < NHEAD; ++ss) acc = fmaf(sw[ss], tokv[ss * HDIM + d], acc);
            xo[d] = acc;
        }
    }
    __syncthreads();

    // ================= Phase 4: proj GEMM + bias + nested output mask ===================
    {
        const int tilebase = wv * 3;       // 48 col tiles over 16 waves
        const float* bp[3];
#pragma unroll
        for (int jj = 0; jj < 3; ++jj)
            bp[jj] = proj_w + (size_t)((tilebase + jj) * 16 + mrow) * DIMX;

        v8f c[3][2] = {};
        const float* ar0 = &qkv[mrow * QSTR];          // attended x (q region)
        const float* ar1 = &qkv[(mrow + 16) * QSTR];
#pragma unroll 2
        for (int kb = 0; kb < DIMX; kb += 4) {
            v2f a0 = *reinterpret_cast<const v2f*>(&ar0[kb + koff]);
            v2f a1 = *reinterpret_cast<const v2f*>(&ar1[kb + koff]);
#pragma unroll
            for (int jj = 0; jj < 3; ++jj) {
                v2f b = *reinterpret_cast<const v2f*>(&bp[jj][kb + koff]);
                c[jj][0] = __builtin_amdgcn_wmma_f32_16x16x4_f32(
                    false, a0, false, b, (short)0, c[jj][0], false, false);
                c[jj][1] = __builtin_amdgcn_wmma_f32_16x16x4_f32(
                    false, a1, false, b, (short)0, c[jj][1], false, false);
            }
        }
#pragma unroll
        for (int jj = 0; jj < 3; ++jj) {
            const int o = (tilebase + jj) * 16 + mrow;  // this lane's output feature
            const float bias = proj_b[o];
#pragma unroll
            for (int rt = 0; rt < 2; ++rt)
#pragma unroll
                for (int i = 0; i < 8; ++i) {
                    const int m = rt * 16 + hi8 + i;
                    const float val = (o < de[m]) ? (c[jj][rt][i] + bias) : 0.0f;
                    out[(size_t)(tb + m) * DIMX + o] = val;
                }
        }
    }
}

extern "C" void kernel_launch(void* const* d_in, const int* in_sizes, int n_in,
                              void* d_out, int out_size, void* d_ws, size_t ws_size,
                              hipStream_t stream) {
    (void)in_sizes; (void)n_in; (void)out_size; (void)d_ws; (void)ws_size;
    const float* x      = (const float*)d_in[0];
    const int*   emask  = (const int*)  d_in[1];
    const float* qkv_w  = (const float*)d_in[2];
    const float* proj_w = (const float*)d_in[3];
    const float* proj_b = (const float*)d_in[4];
    float* out = (float*)d_out;

    const int tokens = 16 * 1024;                        // B * N
    const size_t shmem = (size_t)(2 * TM * XPSTR + TM * QSTR) * sizeof(float); // ~305.5KB
    nested_attn_fused2_kernel<<<tokens / TM, NTHR, shmem, stream>>>(
        x, emask, qkv_w, proj_w, proj_b, out);
}